// LovaszLoss_90469191123020
// MI455X (gfx1250) — compile-verified
//
#include <hip/hip_runtime.h>
#include <hip/hip_bf16.h>

// ---------------------------------------------------------------------------
// Lovasz hinge loss, sort-free rank-histogram formulation for MI455X (gfx1250).
//
//   e_j = 1 - logit_j * (2*t_j - 1);  sorted descending;  result =
//   sum_i relu(e_(i)) * (I - P(i)) / (n - i),   P(i) = positives above i.
//
// Instead of sorting 16M keys (>=1GB radix traffic), bucket the order-
// preserving 32-bit key of e into 2^21 bins (~13 mantissa bits). Each bin is a
// clump sharing (rank r, positives-above P); per-bin sum(relu(e)) is kept in
// u64 fixed point (deterministic integer atomics). Elements with e<=0 never
// influence any used rank (they sort below every contributing element), so
// they are skipped entirely. Max bin population ~700 vs denominators >=2.6M
// where relu>0  => relative error ~1e-4. Traffic ~0.4GB => ~17us @ 23.3TB/s,
// below the full-sort roofline.
//
// CDNA5 specifics: v_wmma_f32_16x16x4_f32 as a wave32 reduction engine
// (all-ones B => D[r][*] = row-sum of A; one WMMA + 7 VALU adds + shfl_xor(16)
// reduces a full wave), global_prefetch_b8 via __builtin_prefetch, wave32
// shuffles. All cross-replay state is re-initialized every launch.
//
// Workspace: 2^21*(4+4+8) + 3*512*4 + 4 bytes ~= 32.01 MB.
// ---------------------------------------------------------------------------

typedef float v2f __attribute__((ext_vector_type(2)));
typedef float v8f __attribute__((ext_vector_type(8)));

#define LOVASZ_NBINS   (1u << 21)     // bins over top-22 key bits; e>0 -> bin < 2^21
#define LOVASZ_SHIFT   10
#define LOVASZ_SCALE   16777216.0f    // 2^24 fixed point for sum(relu(e))
#define LOVASZ_ISCALE  (1.0f / 16777216.0f)
#define K2_BLOCKS      512
#define K2_THREADS     256
#define BINS_PER_TH    16             // 512 * 256 * 16 == 2^21

// Wave32 sum via F32 WMMA: A carries one lane value in slot 0 (zeros
// elsewhere), B is all-ones, so D[r][c] = sum_k A[r][k] regardless of exact
// operand lane layout. Each lane's 8 C VGPRs cover 8 of 16 D rows (low lanes
// rows 0-7, high lanes rows 8-15); summing them plus shfl_xor(16) yields the
// full 32-lane total in every lane. Exact for f32 partials / integer counts.
__device__ __forceinline__ float wmma_wave_sum(float x) {
  v2f a; a[0] = x;    a[1] = 0.0f;
  v2f b; b[0] = 1.0f; b[1] = 1.0f;
  v8f c = {};
  c = __builtin_amdgcn_wmma_f32_16x16x4_f32(false, a, false, b, (short)0, c,
                                            false, false);
  float s = c[0] + c[1] + c[2] + c[3] + c[4] + c[5] + c[6] + c[7];
  s += __shfl_xor(s, 16, 32);   // combine the two 8-row halves (wave32)
  return s;
}

// ---------------------------------------------------------------------------
// K1: build per-bin histograms (count, positive count, fixed-point sum of
// relu(e)) and the global positive total I. Key transform is the standard
// order-preserving float->uint map, inverted so ascending bin == descending e.
// ---------------------------------------------------------------------------
__global__ void __launch_bounds__(256)
lovasz_k1_hist(const float* __restrict__ logits, const int* __restrict__ targets,
               unsigned* __restrict__ cnt, unsigned* __restrict__ pos,
               unsigned long long* __restrict__ sq, unsigned* __restrict__ Icnt,
               int n) {
  __shared__ unsigned sI;
  if (threadIdx.x == 0) sI = 0u;
  __syncthreads();
  unsigned myPos = 0;
  const int stride = gridDim.x * blockDim.x * 4;
  for (int base = (blockIdx.x * blockDim.x + threadIdx.x) * 4; base < n;
       base += stride) {
    if (base + stride < n) {                       // -> global_prefetch_b8
      __builtin_prefetch(logits + base + stride, 0, 0);
      __builtin_prefetch(targets + base + stride, 0, 0);
    }
    const float4 l4 = *(const float4*)(logits + base);
    const int4   t4 = *(const int4*)(targets + base);
#pragma unroll
    for (int j = 0; j < 4; ++j) {
      const float l = (&l4.x)[j];
      const int   t = (&t4.x)[j];
      myPos += (unsigned)(t != 0);
      const float s = t ? 1.0f : -1.0f;
      const float e = fmaf(-s, l, 1.0f);           // 1 - l*sign
      if (e > 0.0f) {                              // e<=0 contributes nothing
        const unsigned u   = __float_as_uint(e);
        const unsigned asc = u ^ ((unsigned)((int)u >> 31) | 0x80000000u);
        const unsigned bin = (~asc) >> LOVASZ_SHIFT;   // < 2^21 when e > 0
        atomicAdd(&cnt[bin], 1u);
        if (t) atomicAdd(&pos[bin], 1u);
        atomicAdd(&sq[bin],
                  (unsigned long long)(e * LOVASZ_SCALE + 0.5f));
      }
    }
  }
  atomicAdd(&sI, myPos);
  __syncthreads();
  if (threadIdx.x == 0) atomicAdd(Icnt, sI);
}

// ---------------------------------------------------------------------------
// K2a: per-block totals of cnt/pos over 4096 contiguous bins (WMMA reduce).
// Counts are < 2^24 so the f32 matrix path is exact.
// ---------------------------------------------------------------------------
__global__ void __launch_bounds__(K2_THREADS)
lovasz_k2a_blocksum(const unsigned* __restrict__ cnt,
                    const unsigned* __restrict__ pos,
                    unsigned* __restrict__ blkCnt, unsigned* __restrict__ blkPos) {
  __shared__ float sred[K2_THREADS / 32];
  const int bin0 =
      blockIdx.x * (K2_THREADS * BINS_PER_TH) + threadIdx.x * BINS_PER_TH;
  unsigned sc = 0, sp = 0;
  const uint4* c4 = (const uint4*)(cnt + bin0);
  const uint4* p4 = (const uint4*)(pos + bin0);
#pragma unroll
  for (int q = 0; q < 4; ++q) {
    uint4 c = c4[q]; sc += c.x + c.y + c.z + c.w;
    uint4 p = p4[q]; sp += p.x + p.y + p.z + p.w;
  }
  const int wid = threadIdx.x >> 5, lane = threadIdx.x & 31;
  float wc = wmma_wave_sum((float)sc);
  if (lane == 0) sred[wid] = wc;
  __syncthreads();
  if (threadIdx.x == 0) {
    float t = 0.0f;
    for (int i = 0; i < K2_THREADS / 32; ++i) t += sred[i];
    blkCnt[blockIdx.x] = (unsigned)(t + 0.5f);
  }
  __syncthreads();
  float wp = wmma_wave_sum((float)sp);
  if (lane == 0) sred[wid] = wp;
  __syncthreads();
  if (threadIdx.x == 0) {
    float t = 0.0f;
    for (int i = 0; i < K2_THREADS / 32; ++i) t += sred[i];
    blkPos[blockIdx.x] = (unsigned)(t + 0.5f);
  }
}

// ---------------------------------------------------------------------------
// K2b: exclusive scan (in place) of the 512 block totals. 1 block, 512 thr.
// ---------------------------------------------------------------------------
__global__ void __launch_bounds__(K2_BLOCKS)
lovasz_k2b_scan(unsigned* __restrict__ blkCnt, unsigned* __restrict__ blkPos) {
  __shared__ unsigned wc[16], wp[16];
  const unsigned c = blkCnt[threadIdx.x], p = blkPos[threadIdx.x];
  unsigned ic = c, ip = p;
  const int lane = threadIdx.x & 31, wid = threadIdx.x >> 5;
#pragma unroll
  for (int o = 1; o < 32; o <<= 1) {
    unsigned uc = __shfl_up(ic, o, 32), up_ = __shfl_up(ip, o, 32);
    if (lane >= o) { ic += uc; ip += up_; }
  }
  if (lane == 31) { wc[wid] = ic; wp[wid] = ip; }
  __syncthreads();
  if (threadIdx.x < 16) {
    unsigned vc = wc[threadIdx.x], vp = wp[threadIdx.x];
#pragma unroll
    for (int o = 1; o < 16; o <<= 1) {
      unsigned uc = __shfl_up(vc, o, 32), up_ = __shfl_up(vp, o, 32);
      if ((int)threadIdx.x >= o) { vc += uc; vp += up_; }
    }
    wc[threadIdx.x] = vc; wp[threadIdx.x] = vp;
  }
  __syncthreads();
  blkCnt[threadIdx.x] = (wid ? wc[wid - 1] : 0u) + (ic - c);  // exclusive
  blkPos[threadIdx.x] = (wid ? wp[wid - 1] : 0u) + (ip - p);
}

// ---------------------------------------------------------------------------
// K2c: per-bin exclusive (r, P) via in-block scan + block offsets; accumulate
// contribution sum(relu(e))_bin * (I - P) / (n - r). WMMA block reduction,
// fixed-order, written to partials[block] (deterministic).
// ---------------------------------------------------------------------------
__global__ void __launch_bounds__(K2_THREADS)
lovasz_k2c_contrib(const unsigned* __restrict__ cnt,
                   const unsigned* __restrict__ pos,
                   const unsigned long long* __restrict__ sq,
                   const unsigned* __restrict__ blkCnt,
                   const unsigned* __restrict__ blkPos,
                   const unsigned* __restrict__ Icnt,
                   float* __restrict__ partials, unsigned n) {
  __shared__ unsigned swc[K2_THREADS / 32], swp[K2_THREADS / 32];
  __shared__ float sred[K2_THREADS / 32];
  const int bin0 =
      blockIdx.x * (K2_THREADS * BINS_PER_TH) + threadIdx.x * BINS_PER_TH;
  unsigned lc[BINS_PER_TH], lp[BINS_PER_TH];
  {
    const uint4* c4 = (const uint4*)(cnt + bin0);
    const uint4* p4 = (const uint4*)(pos + bin0);
#pragma unroll
    for (int q = 0; q < 4; ++q) {
      uint4 c = c4[q];
      lc[4 * q] = c.x; lc[4 * q + 1] = c.y; lc[4 * q + 2] = c.z; lc[4 * q + 3] = c.w;
      uint4 p = p4[q];
      lp[4 * q] = p.x; lp[4 * q + 1] = p.y; lp[4 * q + 2] = p.z; lp[4 * q + 3] = p.w;
    }
  }
  unsigned sc = 0, sp = 0;
#pragma unroll
  for (int j = 0; j < BINS_PER_TH; ++j) { sc += lc[j]; sp += lp[j]; }

  // exclusive scan of per-thread totals across the block (dual u32)
  unsigned ic = sc, ip = sp;
  const int lane = threadIdx.x & 31, wid = threadIdx.x >> 5;
#pragma unroll
  for (int o = 1; o < 32; o <<= 1) {
    unsigned uc = __shfl_up(ic, o, 32), up_ = __shfl_up(ip, o, 32);
    if (lane >= o) { ic += uc; ip += up_; }
  }
  if (lane == 31) { swc[wid] = ic; swp[wid] = ip; }
  __syncthreads();
  if (threadIdx.x < K2_THREADS / 32) {
    unsigned vc = swc[threadIdx.x], vp = swp[threadIdx.x];
#pragma unroll
    for (int o = 1; o < K2_THREADS / 32; o <<= 1) {
      unsigned uc = __shfl_up(vc, o, 32), up_ = __shfl_up(vp, o, 32);
      if ((int)threadIdx.x >= o) { vc += uc; vp += up_; }
    }
    swc[threadIdx.x] = vc; swp[threadIdx.x] = vp;
  }
  __syncthreads();
  unsigned r = blkCnt[blockIdx.x] + (wid ? swc[wid - 1] : 0u) + (ic - sc);
  unsigned P = blkPos[blockIdx.x] + (wid ? swp[wid - 1] : 0u) + (ip - sp);
  const unsigned I = *Icnt;

  float acc = 0.0f;
#pragma unroll
  for (int j = 0; j < BINS_PER_TH; ++j) {
    if (lc[j]) {
      const float sr = (float)sq[bin0 + j] * LOVASZ_ISCALE;  // sum relu(e)
      acc += sr * ((float)(I - P) / (float)(n - r));
    }
    r += lc[j];
    P += lp[j];
  }
  float w = wmma_wave_sum(acc);
  if (lane == 0) sred[wid] = w;
  __syncthreads();
  if (threadIdx.x == 0) {
    float t = 0.0f;
    for (int i = 0; i < K2_THREADS / 32; ++i) t += sred[i];
    partials[blockIdx.x] = t;
  }
}

// ---------------------------------------------------------------------------
// K3: deterministic final reduction of the 512 block partials.
// ---------------------------------------------------------------------------
__global__ void __launch_bounds__(512)
lovasz_k3_final(const float* __restrict__ partials, float* __restrict__ out) {
  __shared__ float sred[16];
  float w = wmma_wave_sum(partials[threadIdx.x]);
  const int wid = threadIdx.x >> 5, lane = threadIdx.x & 31;
  if (lane == 0) sred[wid] = w;
  __syncthreads();
  if (threadIdx.x == 0) {
    float t = 0.0f;
    for (int i = 0; i < 16; ++i) t += sred[i];
    out[0] = t;
  }
}

extern "C" void kernel_launch(void* const* d_in, const int* in_sizes, int n_in,
                              void* d_out, int out_size, void* d_ws,
                              size_t ws_size, hipStream_t stream) {
  const float* logits  = (const float*)d_in[0];
  const int*   targets = (const int*)d_in[1];
  const unsigned n = (unsigned)in_sizes[0];     // 16,777,216 (multiple of 4)

  char* ws = (char*)d_ws;
  size_t off = 0;
  unsigned* cnt = (unsigned*)(ws + off);            off += (size_t)LOVASZ_NBINS * 4;
  unsigned* pos = (unsigned*)(ws + off);            off += (size_t)LOVASZ_NBINS * 4;
  unsigned long long* sq = (unsigned long long*)(ws + off);
                                                    off += (size_t)LOVASZ_NBINS * 8;
  unsigned* blkCnt  = (unsigned*)(ws + off);        off += K2_BLOCKS * 4;
  unsigned* blkPos  = (unsigned*)(ws + off);        off += K2_BLOCKS * 4;
  float*    partials = (float*)(ws + off);          off += K2_BLOCKS * 4;
  unsigned* Icnt    = (unsigned*)(ws + off);        off += 4;
  // requires ws_size >= ~32.01 MB

  hipMemsetAsync(cnt, 0, (size_t)LOVASZ_NBINS * 4, stream);
  hipMemsetAsync(pos, 0, (size_t)LOVASZ_NBINS * 4, stream);
  hipMemsetAsync(sq,  0, (size_t)LOVASZ_NBINS * 8, stream);
  hipMemsetAsync(Icnt, 0, 4, stream);

  lovasz_k1_hist<<<2048, 256, 0, stream>>>(logits, targets, cnt, pos, sq, Icnt,
                                           (int)n);
  lovasz_k2a_blocksum<<<K2_BLOCKS, K2_THREADS, 0, stream>>>(cnt, pos, blkCnt,
                                                            blkPos);
  lovasz_k2b_scan<<<1, K2_BLOCKS, 0, stream>>>(blkCnt, blkPos);
  lovasz_k2c_contrib<<<K2_BLOCKS, K2_THREADS, 0, stream>>>(
      cnt, pos, sq, blkCnt, blkPos, Icnt, partials, n);
  lovasz_k3_final<<<1, 512, 0, stream>>>(partials, (float*)d_out);
}